// SheafHyperGNN_31842887533297
// MI455X (gfx1250) — compile-verified
//
#include <hip/hip_runtime.h>
#include <hip/hip_bf16.h>
#include <math.h>

// ---------------------------------------------------------------------------
// SheafHyperGNN forward for MI455X (gfx1250, wave32).
//  - Dense GEMMs (x@W_lin, xin@W_conv*) use V_WMMA_F32_16X16X4_F32 (full f32:
//    the pipeline is bound by the 1.5M-row gather/scatter traffic, not FLOPs,
//    so there is no payoff for dropping precision).
//  - Incidence scatters use global_atomic_add_f32 (L2 atomics).
//  - Sheaf MLP: one wave32 per incidence, butterfly __shfl_xor reductions.
// Workspace requirement: ~267 MB (66,840,000 floats).
// ---------------------------------------------------------------------------

#define N_NODES 50000
#define N_EDGES 10000
#define NNZ     250000
#define IN_CH   128
#define H_DIM   64
#define D_STALK 6
#define HD      (H_DIM * D_STALK)     // 384
#define LN_LEN  (2 * H_DIM)           // 128

typedef __attribute__((ext_vector_type(2))) float v2f;
typedef __attribute__((ext_vector_type(8))) float v8f;

// ---------------------------------------------------------------------------
// C[M,Nout] = A[M,K] @ B[K,Nout] (+ bias[Nout]).  One wave per 16x16 C tile,
// K stepped by 4 through V_WMMA_F32_16X16X4_F32.  M must be a multiple of 16
// (true for 50000 / 10000 / 300000), Nout a multiple of 16, K of 4.
// A-frag layout (16x4 f32): lane = (k>>1)*16 + m, vgpr = k&1.
// B-frag layout (4x16 f32): lane = (k>>1)*16 + n, vgpr = k&1.
// C layout: vgpr r -> row r (lanes 0-15) / row 8+r (lanes 16-31), col = lane&15.
// ---------------------------------------------------------------------------
__global__ __launch_bounds__(256) void wmma_gemm_bias(
    const float* __restrict__ A, const float* __restrict__ B,
    const float* __restrict__ bias, float* __restrict__ C,
    int M, int K, int Nout) {
  const int lane  = threadIdx.x & 31;
  const int wave  = threadIdx.x >> 5;
  const int tileM = blockIdx.x * 8 + wave;       // 8 waves per block
  if (tileM * 16 >= M) return;                   // wave-uniform: EXEC stays full
  const int row0 = tileM * 16;
  const int col0 = blockIdx.y * 16;
  const int g    = lane >> 4;                    // lane half
  const int mn   = lane & 15;                    // m (A) / n (B,C)

  v8f acc;
  const float bv = bias ? bias[col0 + mn] : 0.0f;  // bias uniform over rows
#pragma unroll
  for (int i = 0; i < 8; ++i) acc[i] = bv;

  const float* arow = A + (size_t)(row0 + mn) * K + 2 * g;
  const float* bcol = B + (size_t)(2 * g) * Nout + col0 + mn;
  for (int k = 0; k < K; k += 4) {
    v2f av, bw;
    av.x = arow[k];
    av.y = arow[k + 1];
    bw.x = bcol[(size_t)k * Nout];
    bw.y = bcol[(size_t)(k + 1) * Nout];
    acc = __builtin_amdgcn_wmma_f32_16x16x4_f32(
        /*neg_a=*/false, av, /*neg_b=*/false, bw,
        /*c_mod=*/(short)0, acc, /*reuse_a=*/false, /*reuse_b=*/false);
  }

  float* crow = C + (size_t)(row0 + g * 8) * Nout + col0 + mn;
#pragma unroll
  for (int r = 0; r < 8; ++r) crow[(size_t)r * Nout] = acc[r];
}

// xa[rows, 64] = mean over the D stalk dims of s[rows*D, 64]
__global__ void stalk_mean(const float* __restrict__ s, float* __restrict__ out,
                           int rows) {
  int t = blockIdx.x * blockDim.x + threadIdx.x;
  if (t >= rows * H_DIM) return;
  int r = t >> 6, c = t & 63;
  const float* p = s + (size_t)r * HD + c;
  float acc = 0.0f;
#pragma unroll
  for (int d = 0; d < D_STALK; ++d) acc += p[d * H_DIM];
  out[t] = acc * (1.0f / (float)D_STALK);
}

// One wave32 per incidence: LN(concat(xa[node], ea[edge])) @ W_sheaf -> sigmoid
__global__ void sheaf_alpha(const float* __restrict__ xa, const float* __restrict__ ea,
                            const int* __restrict__ nidx, const int* __restrict__ eidx,
                            const float* __restrict__ ln_g, const float* __restrict__ ln_b,
                            const float* __restrict__ W_sh, const float* __restrict__ b_sh,
                            float* __restrict__ alpha) {
  int w    = (blockIdx.x * blockDim.x + threadIdx.x) >> 5;
  int lane = threadIdx.x & 31;
  if (w >= NNZ) return;
  int node = nidx[w], edge = eidx[w];
  // lanes 0-15 hold xa[node] (feat 0..63), lanes 16-31 hold ea[edge] (64..127)
  const float* base = (lane < 16) ? (xa + (size_t)node * H_DIM + lane * 4)
                                  : (ea + (size_t)edge * H_DIM + (lane - 16) * 4);
  float f[4];
#pragma unroll
  for (int j = 0; j < 4; ++j) f[j] = base[j];

  float s = f[0] + f[1] + f[2] + f[3];
  float q = f[0]*f[0] + f[1]*f[1] + f[2]*f[2] + f[3]*f[3];
#pragma unroll
  for (int off = 16; off > 0; off >>= 1) {
    s += __shfl_xor(s, off, 32);
    q += __shfl_xor(q, off, 32);
  }
  float mean = s * (1.0f / (float)LN_LEN);
  float var  = q * (1.0f / (float)LN_LEN) - mean * mean;
  float rstd = rsqrtf(var + 1e-5f);

  float acc[D_STALK] = {0, 0, 0, 0, 0, 0};
#pragma unroll
  for (int j = 0; j < 4; ++j) {
    int i = lane * 4 + j;
    float gfe = (f[j] - mean) * rstd * ln_g[i] + ln_b[i];
#pragma unroll
    for (int d = 0; d < D_STALK; ++d) acc[d] += gfe * W_sh[i * D_STALK + d];
  }
#pragma unroll
  for (int off = 16; off > 0; off >>= 1) {
#pragma unroll
    for (int d = 0; d < D_STALK; ++d) acc[d] += __shfl_xor(acc[d], off, 32);
  }
  if (lane == 0) {
#pragma unroll
    for (int d = 0; d < D_STALK; ++d) {
      float z = acc[d] + b_sh[d];
      alpha[(size_t)w * D_STALK + d] = 1.0f / (1.0f + expf(-z));
    }
  }
}

// Degree counts (d-independent: Dn[n*D+d] == deg(node n))
__global__ void degree_count(const int* __restrict__ nidx, const int* __restrict__ eidx,
                             float* __restrict__ degN, float* __restrict__ degE) {
  int t = blockIdx.x * blockDim.x + threadIdx.x;
  if (t >= NNZ) return;
  atomicAdd(&degN[nidx[t]], 1.0f);
  atomicAdd(&degE[eidx[t]], 1.0f);
}

__global__ void invert_deg(float* __restrict__ p, int n) {
  int t = blockIdx.x * blockDim.x + threadIdx.x;
  if (t >= n) return;
  float v = p[t];
  p[t] = (v > 0.0f) ? 1.0f / v : 0.0f;
}

// dst[(sidx[nz]*D+d)*64+col] += alpha[nz,d] * src[(gidx[nz]*D+d)*64+col]
__global__ void scatter_rows(const float* __restrict__ src, const float* __restrict__ alpha,
                             const int* __restrict__ gidx, const int* __restrict__ sidx,
                             float* __restrict__ dst) {
  int t = blockIdx.x * blockDim.x + threadIdx.x;
  if (t >= NNZ * HD) return;                 // 96,000,000 < INT_MAX
  int nz  = t / HD;
  int rem = t - nz * HD;
  int d   = rem >> 6;
  int col = rem & 63;
  float a = alpha[(size_t)nz * D_STALK + d];
  int gi = gidx[nz], si = sidx[nz];
  float v = a * src[((size_t)gi * D_STALK + d) * H_DIM + col];
  atomicAdd(&dst[((size_t)si * D_STALK + d) * H_DIM + col], v);
}

__global__ void row_scale(float* __restrict__ buf, const float* __restrict__ inv,
                          int total) {
  int t = blockIdx.x * blockDim.x + threadIdx.x;
  if (t >= total) return;
  buf[t] *= inv[(t >> 6) / D_STALK];
}

// out = elu(xl - dinv[node]*agg + bias[col])
__global__ void combine_elu(const float* __restrict__ xl, const float* __restrict__ agg,
                            const float* __restrict__ dinv, const float* __restrict__ bias,
                            float* __restrict__ out) {
  int t = blockIdx.x * blockDim.x + threadIdx.x;
  if (t >= N_NODES * HD) return;             // 19,200,000
  int col  = t & 63;
  int node = (t >> 6) / D_STALK;
  float v = xl[t] - dinv[node] * agg[t] + bias[col];
  out[t] = (v > 0.0f) ? v : (expf(v) - 1.0f);
}

extern "C" void kernel_launch(void* const* d_in, const int* in_sizes, int n_in,
                              void* d_out, int out_size, void* d_ws, size_t ws_size,
                              hipStream_t stream) {
  (void)in_sizes; (void)n_in; (void)out_size; (void)ws_size;
  const float* x     = (const float*)d_in[0];
  const float* hattr = (const float*)d_in[1];
  const int*   nidx  = (const int*)d_in[2];
  const int*   eidx  = (const int*)d_in[3];
  // d_in[4] node_types, d_in[5] hyperedge_types: unused by the reference
  const float* W_lin = (const float*)d_in[6];
  const float* b_lin = (const float*)d_in[7];
  const float* ln_g  = (const float*)d_in[8];
  const float* ln_b  = (const float*)d_in[9];
  const float* W_sh  = (const float*)d_in[10];
  const float* b_sh  = (const float*)d_in[11];
  const float* W_c0  = (const float*)d_in[12];
  const float* b_c0  = (const float*)d_in[13];
  const float* W_c1  = (const float*)d_in[14];
  const float* b_c1  = (const float*)d_in[15];
  float* out = (float*)d_out;

  // Workspace carve-up (floats). Total 66,840,000 floats = 267.4 MB.
  float* ws    = (float*)d_ws;
  float* xs    = ws;                        // [300000,64] = 19,200,000
  float* es    = xs + 19200000;             // [ 60000,64] =  3,840,000 (reused as m)
  float* xa    = es + 3840000;              // [ 50000,64] =  3,200,000
  float* ea    = xa + 3200000;              // [ 10000,64] =    640,000
  float* alpha = ea + 640000;               // [250000, 6] =  1,500,000
  float* degN  = alpha + 1500000;           //                  50,000
  float* degE  = degN + 50000;              //                  10,000
  float* xl    = degE + 10000;              // [300000,64] = 19,200,000
  float* agg   = xl + 19200000;             // [300000,64] = 19,200,000
  float* mbuf  = es;                        // alias: es dead before conv0

  const dim3 blk(256);

  // --- input projection: xs = x@W_lin + b_lin ; es = hattr@W_lin + b_lin ---
  wmma_gemm_bias<<<dim3((3125 + 7) / 8, HD / 16), blk, 0, stream>>>(
      x, W_lin, b_lin, xs, N_NODES, IN_CH, HD);
  wmma_gemm_bias<<<dim3((625 + 7) / 8, HD / 16), blk, 0, stream>>>(
      hattr, W_lin, b_lin, es, N_EDGES, IN_CH, HD);

  // --- stalk means ---
  stalk_mean<<<(N_NODES * H_DIM + 255) / 256, blk, 0, stream>>>(xs, xa, N_NODES);
  stalk_mean<<<(N_EDGES * H_DIM + 255) / 256, blk, 0, stream>>>(es, ea, N_EDGES);

  // --- sheaf coefficients ---
  sheaf_alpha<<<(NNZ * 32 + 255) / 256, blk, 0, stream>>>(
      xa, ea, nidx, eidx, ln_g, ln_b, W_sh, b_sh, alpha);

  // --- degree normalizers ---
  hipMemsetAsync(degN, 0, (N_NODES + N_EDGES) * sizeof(float), stream);
  degree_count<<<(NNZ + 255) / 256, blk, 0, stream>>>(nidx, eidx, degN, degE);
  invert_deg<<<(N_NODES + N_EDGES + 255) / 256, blk, 0, stream>>>(degN,
                                                                  N_NODES + N_EDGES);

  // --- two sheaf diffusion convolutions ---
  auto conv = [&](const float* xin, const float* W, const float* bias, float* dst) {
    // xl = xin @ W   (M=300000, K=64, Nout=64)
    wmma_gemm_bias<<<dim3((18750 + 7) / 8, H_DIM / 16), blk, 0, stream>>>(
        xin, W, nullptr, xl, N_NODES * D_STALK, H_DIM, H_DIM);
    // m = Binv * segsum_edge(alpha * xl[node])
    hipMemsetAsync(mbuf, 0, (size_t)N_EDGES * HD * sizeof(float), stream);
    scatter_rows<<<(NNZ * HD + 255) / 256, blk, 0, stream>>>(xl, alpha, nidx, eidx, mbuf);
    row_scale<<<(N_EDGES * HD + 255) / 256, blk, 0, stream>>>(mbuf, degE, N_EDGES * HD);
    // agg = segsum_node(alpha * m[edge])   (Dinv applied in combine)
    hipMemsetAsync(agg, 0, (size_t)N_NODES * HD * sizeof(float), stream);
    scatter_rows<<<(NNZ * HD + 255) / 256, blk, 0, stream>>>(mbuf, alpha, eidx, nidx, agg);
    // dst = elu(xl - Dinv*agg + bias)
    combine_elu<<<(N_NODES * HD + 255) / 256, blk, 0, stream>>>(xl, agg, degN, bias, dst);
  };

  conv(xs, W_c0, b_c0, xs);   // h1 = elu(conv0(xs)), overwrites xs
  conv(xs, W_c1, b_c1, out);  // out = elu(conv1(h1))
}